// AttentiveLayer_73469710566058
// MI455X (gfx1250) — compile-verified
//
#include <hip/hip_runtime.h>

typedef __attribute__((ext_vector_type(4))) float v4f;
typedef __attribute__((ext_vector_type(4))) int v4i;

#define THREADS 256
#define ROWS_PER_ITER 4

__global__ __launch_bounds__(THREADS) void colscale_nt_kernel(
    const float* __restrict__ x, const float* __restrict__ w,
    float* __restrict__ out, int n_rows, int d) {
  // d == 4 * THREADS == 1024 for this problem; thread t owns columns [4t, 4t+3].
  __shared__ float ws[4 * THREADS];
  const int t = threadIdx.x;

  // ---- Stage w (4 KB) into LDS once per block via gfx1250 async DMA ----
#if defined(__AMDGCN__) && __has_builtin(__builtin_amdgcn_global_load_async_to_lds_b128)
  {
    typedef __attribute__((address_space(1))) v4i gv4i;  // prints as "__device__"
    typedef __attribute__((address_space(3))) v4i lv4i;  // prints as "__shared__"
    // Global generic addresses are numerically identical to AS1; generic
    // shared addresses carry the LDS offset in the low 32 bits (ISA 10.2).
    gv4i* gp = (gv4i*)(unsigned long long)(const void*)(w + 4 * t);
    lv4i* lp = (lv4i*)(unsigned)(unsigned long long)(void*)(&ws[4 * t]);
    __builtin_amdgcn_global_load_async_to_lds_b128(gp, lp, 0, 0);
#if __has_builtin(__builtin_amdgcn_s_wait_asynccnt)
    __builtin_amdgcn_s_wait_asynccnt(0);
#else
    asm volatile("s_wait_asynccnt 0" ::: "memory");
#endif
  }
#else
  *(v4f*)(&ws[4 * t]) = *(const v4f*)(w + 4 * t);
#endif
  __syncthreads();

  // Per-thread column scale, held in 4 VGPRs for the whole kernel.
  const v4f wv = *(const v4f*)(&ws[4 * t]);

  const long long row_stride = (long long)gridDim.x * ROWS_PER_ITER;
  const long long elem_stride = row_stride * d;

  for (long long r0 = (long long)blockIdx.x * ROWS_PER_ITER; r0 < n_rows;
       r0 += row_stride) {
    const float* xp = x + r0 * d + 4 * t;
    float* op = out + r0 * d + 4 * t;

    // Speculative prefetch of next grid-stride chunk (global_prefetch_b8);
    // OOB translation failures are silently dropped per ISA 10.5.
    __builtin_prefetch(xp + elem_stride, 0, 1);

    // 4 independent 128-bit non-temporal loads -> MLP; streaming working set
    // (1 GB) >> L2 (192 MB), so NT keeps it from thrashing the caches.
    v4f a0 = __builtin_nontemporal_load((const v4f*)(xp));
    v4f a1 = __builtin_nontemporal_load((const v4f*)(xp + d));
    v4f a2 = __builtin_nontemporal_load((const v4f*)(xp + 2 * d));
    v4f a3 = __builtin_nontemporal_load((const v4f*)(xp + 3 * d));

    a0 *= wv;
    a1 *= wv;
    a2 *= wv;
    a3 *= wv;

    __builtin_nontemporal_store(a0, (v4f*)(op));
    __builtin_nontemporal_store(a1, (v4f*)(op + d));
    __builtin_nontemporal_store(a2, (v4f*)(op + 2 * d));
    __builtin_nontemporal_store(a3, (v4f*)(op + 3 * d));
  }
}

extern "C" void kernel_launch(void* const* d_in, const int* in_sizes, int n_in,
                              void* d_out, int out_size, void* d_ws, size_t ws_size,
                              hipStream_t stream) {
  const float* x = (const float*)d_in[0];
  const float* w = (const float*)d_in[1];
  float* out = (float*)d_out;

  const int d = in_sizes[1];            // 1024
  const int n_rows = in_sizes[0] / d;   // 131072

  dim3 grid(4096);
  dim3 block(THREADS);
  hipLaunchKernelGGL(colscale_nt_kernel, grid, block, 0, stream, x, w, out,
                     n_rows, d);
}